// RNN_7902739825376
// MI455X (gfx1250) — compile-verified
//
#include <hip/hip_runtime.h>

// Problem dims (fixed by the reference)
#define AA   8
#define BBS  512
#define TTS  256
#define VVS  128
#define HHS  64
#define NACT 32
#define NNS  (AA * BBS)   // 4096 independent sequences

typedef float v2f __attribute__((ext_vector_type(2)));
typedef float v8f __attribute__((ext_vector_type(8)));

// D(16x16,f32) = A(16x4,f32) * B(4x16,f32) + C   -- CDNA5 fp32 WMMA
__device__ __forceinline__ v8f wmma4(v2f a, v2f b, v8f c) {
  return __builtin_amdgcn_wmma_f32_16x16x4_f32(
      /*neg_a=*/false, a, /*neg_b=*/false, b,
      /*c_mod=*/(short)0, c, /*reuse_a=*/false, /*reuse_b=*/false);
}

// CDNA5 async DMA: copies 16B per lane from global address g to LDS byte offset lds_byte_off, ASYNCcnt-tracked
__device__ __forceinline__ void async_ld_b128(unsigned lds_byte_off, const float* g) {
  asm volatile("global_load_async_to_lds_b128 %0, %1, off"
               :: "v"(lds_byte_off), "v"(g) : "memory");
}
__device__ __forceinline__ void wait_async0() {
  asm volatile("s_wait_asynccnt 0x0" ::: "memory");
}

// A-matrix 16x4 fragment from a row-major LDS tile [16][stride]:
//   lanes 0-15: VGPR0=K0, VGPR1=K1 ; lanes 16-31: VGPR0=K2, VGPR1=K3
__device__ __forceinline__ v2f ld_afrag(const float* Ls, int stride, int kk, int lane) {
  const int col  = lane & 15;
  const int half = lane >> 4;
  return *(const v2f*)(Ls + col * stride + kk * 4 + half * 2);
}

// B-matrix 4x16 fragment pre-swizzled into contiguous 256B blobs per (kstep, ntile)
__device__ __forceinline__ v2f ld_bfrag(const float* Bf, int f, int lane) {
  return *(const v2f*)(Bf + ((f * 32 + lane) << 1));
}

// Pre-swizzle W[N, K] (used as B = W^T, K x N) into fragment blobs:
// Bf[((kk*NT + nt)*32 + lane)*2 + j] = W[(nt*16 + col)*K + 4*kk + 2*half + j]
__device__ __forceinline__ void fill_bfrag(float* Bf, const float* W,
                                           int KSTEPS, int NT, int K, int tid) {
  const int total = KSTEPS * NT * 32;
  for (int i = tid; i < total; i += 128) {
    const int lane = i & 31;
    const int f    = i >> 5;
    const int nt   = f % NT;
    const int kk   = f / NT;
    const int col  = lane & 15;
    const int half = lane >> 4;
    const int row  = nt * 16 + col;
    const int k0   = kk * 4 + half * 2;
    Bf[i * 2 + 0] = W[row * K + k0 + 0];
    Bf[i * 2 + 1] = W[row * K + k0 + 1];
  }
}

#define XS_STRIDE 132   // 128 + 4 pad -> bank stride 4 mod 64 (conflict free b64 frags)
#define HS_STRIDE 68    // 64 + 4 pad

// LDS layout offsets (floats), dynamic smem starts at LDS offset 0
#define OFF_BF_ENC 0
#define OFF_BF_IH  (OFF_BF_ENC + 32 * 4 * 64)    // 8192
#define OFF_BF_HH  (OFF_BF_IH  + 16 * 12 * 64)   // 20480
#define OFF_BF_OUT (OFF_BF_HH  + 16 * 12 * 64)   // 32768
#define OFF_XS     (OFF_BF_OUT + 16 * 2 * 64)    // 34816
#define OFF_ENC    (OFF_XS     + 16 * XS_STRIDE) // 36928
#define OFF_HS     (OFF_ENC    + 16 * HS_STRIDE) // 38016
#define OFF_BE     (OFF_HS     + 16 * HS_STRIDE) // 39104
#define OFF_BI     (OFF_BE + 64)
#define OFF_BH     (OFF_BI + 192)
#define OFF_BO     (OFF_BH + 192)
#define SMEM_FLOATS (OFF_BO + 32)

__global__ __launch_bounds__(128) void gru_fused_kernel(
    const float* __restrict__ x,     const float* __restrict__ h0,
    const float* __restrict__ W_enc, const float* __restrict__ b_enc,
    const float* __restrict__ W_ih,  const float* __restrict__ b_ih,
    const float* __restrict__ W_hh,  const float* __restrict__ b_hh,
    const float* __restrict__ W_out, const float* __restrict__ b_out,
    float* __restrict__ outp,        float* __restrict__ hidp) {
  extern __shared__ float smem[];
  float* Bf_enc = smem + OFF_BF_ENC;
  float* Bf_ih  = smem + OFF_BF_IH;
  float* Bf_hh  = smem + OFF_BF_HH;
  float* Bf_out = smem + OFF_BF_OUT;
  float* xs     = smem + OFF_XS;     // [16][132]
  float* enc_s  = smem + OFF_ENC;    // [16][68]
  float* h_s    = smem + OFF_HS;     // [16][68]
  float* bias_e = smem + OFF_BE;
  float* bias_i = smem + OFF_BI;
  float* bias_h = smem + OFF_BH;
  float* bias_o = smem + OFF_BO;

  const int tid  = threadIdx.x;
  const int lane = tid & 31;
  const int w    = tid >> 5;          // wave id 0..3 -> owns H-cols 16w..16w+15
  const int col  = lane & 15;
  const int half = lane >> 4;
  const int n0   = blockIdx.x * 16;   // first batch row of this tile
  const int gcol = w * 16 + col;      // this lane's H column

  // ---- async x-tile DMA setup: wave w owns rows 4w..4w+3, lane covers 16B ----
  const float* gsrc[4];
  unsigned     ldst[4];
#pragma unroll
  for (int i = 0; i < 4; ++i) {
    const int row = 4 * w + i;
    gsrc[i] = x + ((size_t)(n0 + row) * TTS + 0) * VVS + lane * 4;
    ldst[i] = (unsigned)((OFF_XS + row * XS_STRIDE + lane * 4) * sizeof(float));
  }
  // kick off tile t=0, overlapped with the weight swizzle below
#pragma unroll
  for (int i = 0; i < 4; ++i) { async_ld_b128(ldst[i], gsrc[i]); gsrc[i] += VVS; }

  // ---- one-time setup: swizzle weights + biases + h0 into LDS ----
  fill_bfrag(Bf_enc, W_enc, 32, 4, VVS, tid);
  fill_bfrag(Bf_ih,  W_ih,  16, 12, HHS, tid);
  fill_bfrag(Bf_hh,  W_hh,  16, 12, HHS, tid);
  fill_bfrag(Bf_out, W_out, 16, 2,  HHS, tid);
  for (int i = tid; i < 64; i += 128)  bias_e[i] = b_enc[i];
  for (int i = tid; i < 192; i += 128) { bias_i[i] = b_ih[i]; bias_h[i] = b_hh[i]; }
  if (tid < 32) bias_o[tid] = b_out[tid];
  for (int i = tid; i < 16 * 64; i += 128) {
    const int r = i >> 6, h = i & 63;
    h_s[r * HS_STRIDE + h] = h0[(size_t)(n0 + r) * HHS + h];
  }
  wait_async0();      // tile 0 resident in xs
  __syncthreads();

  for (int t = 0; t < TTS; ++t) {
    // ---- enc = relu(x @ W_enc^T + b_enc); xs is valid (wait+barrier above) ----
    v8f acc = {};
#pragma unroll
    for (int kk = 0; kk < 32; ++kk) {
      acc = wmma4(ld_afrag(xs, XS_STRIDE, kk, lane),
                  ld_bfrag(Bf_enc, kk * 4 + w, lane), acc);
    }
    {
      const float be = bias_e[gcol];
#pragma unroll
      for (int j = 0; j < 8; ++j) {
        const float v = acc[j] + be;
        enc_s[(j + 8 * half) * HS_STRIDE + gcol] = v > 0.f ? v : 0.f;
      }
    }
    __syncthreads();   // all waves done reading xs -> safe to overwrite via DMA

    // ---- issue async DMA for tile t+1; completes under the GEMMs below ----
    if (t + 1 < TTS) {
#pragma unroll
      for (int i = 0; i < 4; ++i) { async_ld_b128(ldst[i], gsrc[i]); gsrc[i] += VVS; }
    }

    // ---- gx = enc @ W_ih^T ; gh = h @ W_hh^T (3 gate tiles each per wave) ----
    v8f gxr = {}, gxz = {}, gxn = {};
#pragma unroll
    for (int kk = 0; kk < 16; ++kk) {
      const v2f a = ld_afrag(enc_s, HS_STRIDE, kk, lane);
      gxr = wmma4(a, ld_bfrag(Bf_ih, kk * 12 + 0 + w, lane), gxr);
      gxz = wmma4(a, ld_bfrag(Bf_ih, kk * 12 + 4 + w, lane), gxz);
      gxn = wmma4(a, ld_bfrag(Bf_ih, kk * 12 + 8 + w, lane), gxn);
    }
    v8f ghr = {}, ghz = {}, ghn = {};
#pragma unroll
    for (int kk = 0; kk < 16; ++kk) {
      const v2f a = ld_afrag(h_s, HS_STRIDE, kk, lane);
      ghr = wmma4(a, ld_bfrag(Bf_hh, kk * 12 + 0 + w, lane), ghr);
      ghz = wmma4(a, ld_bfrag(Bf_hh, kk * 12 + 4 + w, lane), ghz);
      ghn = wmma4(a, ld_bfrag(Bf_hh, kk * 12 + 8 + w, lane), ghn);
    }
    __syncthreads();  // all waves finished reading h_s / enc_s

    // ---- gates + hidden update (trans VALU co-executes with XDL) ----
    {
      const float bxr = bias_i[gcol], bxz = bias_i[64 + gcol], bxn = bias_i[128 + gcol];
      const float bhr = bias_h[gcol], bhz = bias_h[64 + gcol], bhn = bias_h[128 + gcol];
#pragma unroll
      for (int j = 0; j < 8; ++j) {
        const int r = j + 8 * half;
        const float hold = h_s[r * HS_STRIDE + gcol];
        const float rr = 1.f / (1.f + __expf(-(gxr[j] + bxr + ghr[j] + bhr)));
        const float zz = 1.f / (1.f + __expf(-(gxz[j] + bxz + ghz[j] + bhz)));
        const float pn = gxn[j] + bxn + rr * (ghn[j] + bhn);
        const float e2 = __expf(2.f * pn);
        const float nn = (e2 - 1.f) / (e2 + 1.f);   // tanh
        const float hn = (1.f - zz) * nn + zz * hold;
        h_s[r * HS_STRIDE + gcol] = hn;
        hidp[((size_t)(n0 + r) * TTS + t) * HHS + gcol] = hn;
      }
    }
    __syncthreads();

    // ---- out = h_new @ W_out^T + b_out (2 tiles, waves 0-1) ----
    if (w < 2) {
      v8f oacc = {};
#pragma unroll
      for (int kk = 0; kk < 16; ++kk) {
        oacc = wmma4(ld_afrag(h_s, HS_STRIDE, kk, lane),
                     ld_bfrag(Bf_out, kk * 2 + w, lane), oacc);
      }
      const float bo = bias_o[w * 16 + col];
#pragma unroll
      for (int j = 0; j < 8; ++j) {
        outp[((size_t)(n0 + j + 8 * half) * TTS + t) * NACT + w * 16 + col] =
            oacc[j] + bo;
      }
    }

    wait_async0();     // tile t+1 resident before anyone reads xs next iteration
    __syncthreads();
  }
}

extern "C" void kernel_launch(void* const* d_in, const int* in_sizes, int n_in,
                              void* d_out, int out_size, void* d_ws, size_t ws_size,
                              hipStream_t stream) {
  (void)in_sizes; (void)n_in; (void)out_size; (void)d_ws; (void)ws_size;
  const float* x     = (const float*)d_in[0];
  const float* h0    = (const float*)d_in[1];
  const float* W_enc = (const float*)d_in[2];
  const float* b_enc = (const float*)d_in[3];
  const float* W_ih  = (const float*)d_in[4];
  const float* b_ih  = (const float*)d_in[5];
  const float* W_hh  = (const float*)d_in[6];
  const float* b_hh  = (const float*)d_in[7];
  const float* W_out = (const float*)d_in[8];
  const float* b_out = (const float*)d_in[9];

  float* outp = (float*)d_out;                          // [A,B,T,32]
  float* hidp = outp + (size_t)NNS * TTS * NACT;        // [A,B,T,64]

  const size_t smem_bytes = (size_t)SMEM_FLOATS * sizeof(float);  // ~155 KB

  gru_fused_kernel<<<dim3(NNS / 16), dim3(128), smem_bytes, stream>>>(
      x, h0, W_enc, b_enc, W_ih, b_ih, W_hh, b_hh, W_out, b_out, outp, hidp);
}